// SphericalChebBNPool_46540265619868
// MI455X (gfx1250) — compile-verified
//
#include <hip/hip_runtime.h>
#include <hip/hip_bf16.h>

// ---------------- problem constants ----------------
#define BATCH 16
#define NV    49152
#define CIN   32
#define COUT  64
#define KCH   4            // Chebyshev order
#define NNZ   (NV * 9)
#define POOLF 4
#define FEAT  (BATCH * CIN)          // 512 = SpMM feature width
#define MROWS ((size_t)NV * BATCH)   // 786432 GEMM rows
#define KKDIM (KCH * CIN)            // 128 GEMM reduction dim
#define CHEBN ((size_t)MROWS * CIN)  // floats per Chebyshev term

typedef __attribute__((ext_vector_type(2))) float v2f;
typedef __attribute__((ext_vector_type(8))) float v8f;

// ---------------- kernel 1: zero stats ----------------
__global__ void zero_stats_kernel(float* __restrict__ p) {
    if (threadIdx.x < 128) p[threadIdx.x] = 0.0f;
}

// ---------------- kernel 2: x [B,V,Cin] -> cheb0 [V,B,Cin] ----------------
__global__ void __launch_bounds__(256) transpose_kernel(const float* __restrict__ x,
                                                        float* __restrict__ x0) {
    size_t idx = (size_t)blockIdx.x * blockDim.x + threadIdx.x;
    if (idx >= (size_t)BATCH * NV * CIN) return;
    int c = (int)(idx % CIN);
    size_t t = idx / CIN;
    int v = (int)(t % NV);
    int b = (int)(t / NV);
    x0[((size_t)v * BATCH + b) * CIN + c] = x[idx];
}

// ---------------- kernel 3: CSR row pointers by binary search ----------------
__global__ void rowptr_kernel(const long long* __restrict__ rows, int* __restrict__ rp) {
    int v = blockIdx.x * blockDim.x + threadIdx.x;
    if (v > NV) return;
    int lo = 0, hi = NNZ;
    long long key = (long long)v;
    while (lo < hi) {
        int mid = (lo + hi) >> 1;
        if (rows[mid] < key) lo = mid + 1; else hi = mid;
    }
    rp[v] = lo;   // rp[NV] == NNZ since all rows < NV
}

// ---------------- kernel 4: Xout = alpha*(L @ Xin) + beta*Xprev ----------------
__global__ void __launch_bounds__(128) spmm_kernel(const float* __restrict__ Xin,
                                                   const float* __restrict__ Xprev,
                                                   float* __restrict__ Xout,
                                                   const long long* __restrict__ cols,
                                                   const float* __restrict__ vals,
                                                   const int* __restrict__ rp,
                                                   float alpha, float beta) {
    int v = blockIdx.x;
    int f = threadIdx.x * 4;                 // 128 threads * float4 = 512 features
    int s = rp[v], e = rp[v + 1];
    float ax = 0.f, ay = 0.f, az = 0.f, aw = 0.f;
    for (int j = s; j < e; ++j) {
        float val = vals[j];
        size_t col = (size_t)cols[j];
        const float4 xr = *(const float4*)(Xin + col * FEAT + f);
        ax = fmaf(val, xr.x, ax);
        ay = fmaf(val, xr.y, ay);
        az = fmaf(val, xr.z, az);
        aw = fmaf(val, xr.w, aw);
    }
    float4 o;
    if (beta != 0.0f) {
        const float4 p = *(const float4*)(Xprev + (size_t)v * FEAT + f);
        o.x = fmaf(alpha, ax, beta * p.x);
        o.y = fmaf(alpha, ay, beta * p.y);
        o.z = fmaf(alpha, az, beta * p.z);
        o.w = fmaf(alpha, aw, beta * p.w);
    } else {
        o.x = alpha * ax; o.y = alpha * ay; o.z = alpha * az; o.w = alpha * aw;
    }
    *(float4*)(Xout + (size_t)v * FEAT + f) = o;
}

// ---------------- kernel 5: WMMA GEMM  y[m,o] = sum_kk cheb[m,kk]*W[kk,o]
//                  + fused BatchNorm partial statistics ----------------
// LDS holds W interleaved as K-row pairs: ldsWp[p*64+o] = {W[2p][o], W[2p+1][o]},
// so each WMMA B operand is one aligned ds_load_b64 (no repacking movs).
__global__ void __launch_bounds__(256) gemm_bn_stats_kernel(const float* __restrict__ cheb,
                                                            const float* __restrict__ w,
                                                            float* __restrict__ y,
                                                            float* __restrict__ gsum,
                                                            float* __restrict__ gsumsq) {
    __shared__ v2f   ldsWp[(KKDIM / 2) * COUT];   // 64 pairs x 64 ch = 32 KB
    __shared__ float ldsSum[COUT];
    __shared__ float ldsSq[COUT];

    const int tid = threadIdx.x;
    for (int i = tid; i < (KKDIM / 2) * COUT; i += 256) {
        int p = i >> 6;          // K-row pair index (0..63)
        int o = i & 63;          // output channel
        v2f t;
        t.x = w[(2 * p)     * COUT + o];
        t.y = w[(2 * p + 1) * COUT + o];
        ldsWp[i] = t;
    }
    if (tid < COUT) { ldsSum[tid] = 0.0f; ldsSq[tid] = 0.0f; }
    __syncthreads();

    const int wave = tid >> 5;
    const int lane = tid & 31;
    const int half = lane >> 4;     // 0: lanes 0-15, 1: lanes 16-31
    const int l16  = lane & 15;
    const size_t row0 = ((size_t)blockIdx.x * 8 + wave) * 16;
    const size_t m    = row0 + l16;

    v8f acc0 = {}, acc1 = {}, acc2 = {}, acc3 = {};

    #pragma unroll
    for (int kt = 0; kt < 32; ++kt) {            // 128 / 4
        const int kterm = kt >> 3;               // Chebyshev term
        const int ci    = (kt & 7) * 4 + 2 * half;
        const float* ap = cheb + (size_t)kterm * CHEBN + m * CIN + ci;
        v2f a;
        a.x = ap[0];
        a.y = ap[1];
        // K-row pair index for this step/half: (kt*4 + 2*half)/2 = kt*2 + half
        const v2f* wp = ldsWp + (kt * 2 + half) * COUT;
        v2f b0 = wp[ 0 + l16];
        v2f b1 = wp[16 + l16];
        v2f b2 = wp[32 + l16];
        v2f b3 = wp[48 + l16];
        acc0 = __builtin_amdgcn_wmma_f32_16x16x4_f32(false, a, false, b0, (short)0, acc0, false, false);
        acc1 = __builtin_amdgcn_wmma_f32_16x16x4_f32(false, a, false, b1, (short)0, acc1, false, false);
        acc2 = __builtin_amdgcn_wmma_f32_16x16x4_f32(false, a, false, b2, (short)0, acc2, false, false);
        acc3 = __builtin_amdgcn_wmma_f32_16x16x4_f32(false, a, false, b3, (short)0, acc3, false, false);
    }

    // Epilogue: store y tile + per-channel partial sum / sum-of-squares
    {
        float s1, s2;
        #define EPI(ACC, NT)                                                         \
            s1 = 0.0f; s2 = 0.0f;                                                    \
            _Pragma("unroll")                                                        \
            for (int r = 0; r < 8; ++r) {                                            \
                float d = ACC[r];                                                    \
                y[(row0 + (size_t)r + 8 * half) * COUT + (NT) * 16 + l16] = d;       \
                s1 += d; s2 += d * d;                                                \
            }                                                                        \
            atomicAdd(&ldsSum[(NT) * 16 + l16], s1);                                 \
            atomicAdd(&ldsSq [(NT) * 16 + l16], s2);
        EPI(acc0, 0)
        EPI(acc1, 1)
        EPI(acc2, 2)
        EPI(acc3, 3)
        #undef EPI
    }
    __syncthreads();
    if (tid < COUT)            atomicAdd(&gsum[tid], ldsSum[tid]);
    else if (tid < 2 * COUT)   atomicAdd(&gsumsq[tid - COUT], ldsSq[tid - COUT]);
}

// ---------------- kernel 6: finalize BN scale/shift ----------------
__global__ void finalize_stats_kernel(const float* __restrict__ gsum,
                                      const float* __restrict__ gsumsq,
                                      const float* __restrict__ gamma,
                                      const float* __restrict__ beta,
                                      float* __restrict__ scale,
                                      float* __restrict__ shift) {
    int c = threadIdx.x;
    if (c >= COUT) return;
    const float n = (float)((size_t)BATCH * NV);
    float mean = gsum[c] / n;
    float var  = gsumsq[c] / n - mean * mean;
    float sc   = gamma[c] * rsqrtf(var + 1e-5f);
    scale[c] = sc;
    shift[c] = beta[c] - mean * sc;
}

// ---------------- kernel 7: BN + ReLU + 4:1 vertex pooling ----------------
__global__ void __launch_bounds__(256) bn_pool_kernel(const float* __restrict__ y,
                                                      const float* __restrict__ scale,
                                                      const float* __restrict__ shift,
                                                      float* __restrict__ out) {
    size_t idx = (size_t)blockIdx.x * blockDim.x + threadIdx.x;
    if (idx >= (size_t)BATCH * (NV / POOLF) * COUT) return;
    int o = (int)(idx % COUT);
    size_t t = idx / COUT;
    int vp = (int)(t % (NV / POOLF));
    int b  = (int)(t / (NV / POOLF));
    const float sc = scale[o];
    const float sh = shift[o];
    float acc = 0.0f;
    #pragma unroll
    for (int p = 0; p < POOLF; ++p) {
        size_t row = (size_t)(vp * POOLF + p) * BATCH + b;   // y row index = v*B + b
        float val = fmaf(y[row * COUT + o], sc, sh);
        acc += fmaxf(val, 0.0f);
    }
    out[idx] = acc * 0.25f;
}

// ---------------- host launcher ----------------
extern "C" void kernel_launch(void* const* d_in, const int* in_sizes, int n_in,
                              void* d_out, int out_size, void* d_ws, size_t ws_size,
                              hipStream_t stream) {
    const float*     x        = (const float*)d_in[0];
    const float*     lap_vals = (const float*)d_in[1];
    const float*     weight   = (const float*)d_in[2];
    const float*     gamma    = (const float*)d_in[3];
    const float*     beta     = (const float*)d_in[4];
    const long long* lap_rows = (const long long*)d_in[5];
    const long long* lap_cols = (const long long*)d_in[6];
    float* out = (float*)d_out;

    // workspace layout
    char* ws = (char*)d_ws;
    float* gsum   = (float*)ws;            // 64
    float* gsumsq = gsum + COUT;           // 64
    float* scale  = gsumsq + COUT;         // 64
    float* shift  = scale + COUT;          // 64
    int*   rp     = (int*)(ws + 1024);     // NV+1 ints
    size_t off = 1024 + sizeof(int) * (size_t)(NV + 1);
    off = (off + 511) & ~(size_t)511;
    float* cheb = (float*)(ws + off);      // 4 contiguous terms, CHEBN floats each
    float* yb   = cheb + 4 * CHEBN;        // MROWS * COUT floats

    float* c0 = cheb + 0 * CHEBN;
    float* c1 = cheb + 1 * CHEBN;
    float* c2 = cheb + 2 * CHEBN;
    float* c3 = cheb + 3 * CHEBN;

    zero_stats_kernel<<<1, 128, 0, stream>>>(gsum);

    {
        size_t tot = (size_t)BATCH * NV * CIN;
        transpose_kernel<<<(unsigned)((tot + 255) / 256), 256, 0, stream>>>(x, c0);
    }
    rowptr_kernel<<<(NV + 1 + 255) / 256, 256, 0, stream>>>(lap_rows, rp);

    // Chebyshev recurrence: X1 = L X0 ; X2 = 2 L X1 - X0 ; X3 = 2 L X2 - X1
    spmm_kernel<<<NV, 128, 0, stream>>>(c0, c0, c1, lap_cols, lap_vals, rp, 1.0f,  0.0f);
    spmm_kernel<<<NV, 128, 0, stream>>>(c1, c0, c2, lap_cols, lap_vals, rp, 2.0f, -1.0f);
    spmm_kernel<<<NV, 128, 0, stream>>>(c2, c1, c3, lap_cols, lap_vals, rp, 2.0f, -1.0f);

    // GEMM (786432 x 128) @ (128 x 64) with fused BN statistics
    gemm_bn_stats_kernel<<<(unsigned)(MROWS / 16 / 8), 256, 0, stream>>>(cheb, weight, yb,
                                                                         gsum, gsumsq);

    finalize_stats_kernel<<<1, 64, 0, stream>>>(gsum, gsumsq, gamma, beta, scale, shift);

    {
        size_t tot = (size_t)BATCH * (NV / POOLF) * COUT;
        bn_pool_kernel<<<(unsigned)((tot + 255) / 256), 256, 0, stream>>>(yb, scale, shift, out);
    }
}